// AttentionSymGAT_30709016166929
// MI455X (gfx1250) — compile-verified
//
#include <hip/hip_runtime.h>
#include <hip/hip_bf16.h>
#include <float.h>

typedef __attribute__((ext_vector_type(2))) float v2f;
typedef __attribute__((ext_vector_type(8))) float v8f;

#define B_  2
#define NN_ 50000
#define E_  800000
#define H_  8
#define D_  128

// ---------------------------------------------------------------------------
// Kernel 1: attention scores via V_WMMA_F32_16X16X4_F32 (exact fp32 path).
// One wave per tile of 2 nodes = 16 contiguous (n,h) rows x D=128 (8 KB).
// B (4x16) columns: 0..7 = ks head h, 8..15 = ka head h; 32 chained WMMAs.
// Verified codegen: b64 loads pipelined 9-deep against the WMMA chain.
// ---------------------------------------------------------------------------
__global__ void k_attn_scores(const float* __restrict__ X,
                              const float* __restrict__ ks,
                              const float* __restrict__ ka,
                              float* __restrict__ AS,
                              float* __restrict__ AA,
                              int numTiles) {
  const int lane   = threadIdx.x & 31;
  const int wave   = blockIdx.x * (blockDim.x >> 5) + (threadIdx.x >> 5);
  const int nWaves = gridDim.x * (blockDim.x >> 5);
  const int rc     = lane & 15;           // A: row within tile, B: column
  const int kHalf  = (lane >> 4) << 1;    // 0 or 2

  // Preload B-matrix chunks (shared by every tile): ks/ka stored as (D,H,1).
  const float* P    = (rc < 8) ? ks : ka;
  const int    hcol = rc & 7;
  v2f bmat[32];
#pragma unroll
  for (int kk = 0; kk < 32; ++kk) {
    const int k0 = kk * 4 + kHalf;
    bmat[kk].x = P[k0 * H_ + hcol];
    bmat[kk].y = P[(k0 + 1) * H_ + hcol];
  }

  const int node  = lane >> 4;        // which of the 2 nodes this lane extracts
  const int isAdj = (lane >> 3) & 1;  // lanes 8-15 / 24-31 hold adjc columns

  for (int t = wave; t < numTiles; t += nWaves) {
    const float* rowPtr = X + (size_t)t * (2 * H_ * D_) + rc * D_ + kHalf;
    v8f c = {0.f, 0.f, 0.f, 0.f, 0.f, 0.f, 0.f, 0.f};
#pragma unroll
    for (int kk = 0; kk < 32; ++kk) {
      v2f a;
      a.x = rowPtr[kk * 4];
      a.y = rowPtr[kk * 4 + 1];
      c = __builtin_amdgcn_wmma_f32_16x16x4_f32(false, a, false, bmat[kk],
                                                (short)0, c, false, false);
    }
    // Useful outputs of C: (M = ln*8+h, N = h) -> self, (M, N = 8+h) -> adjc.
    float* dst = isAdj ? AA : AS;
    const size_t gbase = ((size_t)t * 2 + node) * H_;
#pragma unroll
    for (int r = 0; r < 8; ++r) {
      if ((lane & 7) == r) dst[gbase + r] = c[r];
    }
  }
}

// ---------------------------------------------------------------------------
// Init m / m2 to -inf (only non-empty segments are ever read back).
// ---------------------------------------------------------------------------
__global__ void k_init(float* __restrict__ p, int n) {
  int i = blockIdx.x * blockDim.x + threadIdx.x;
  if (i < n) p[i] = -FLT_MAX;
}

// ---------------------------------------------------------------------------
// Recompute-instead-of-spill helper: 4-way gather (incl. double indirection)
// + leaky_relu(0.2). All gather sources (AS/AA 6.4MB, targets/sources 12.8MB)
// are L2-resident, so recomputing per phase beats streaming a 51MB coef
// buffer through HBM three extra times.
// ---------------------------------------------------------------------------
__device__ __forceinline__ int gather_coef(const float* __restrict__ AS,
                                           const float* __restrict__ AA,
                                           const int* __restrict__ targets,
                                           const int* __restrict__ sources,
                                           int i, float v[H_]) {
  const int b = i / E_;
  const int t = targets[i];
  const int s = sources[i];
  const int t_of_s = targets[b * E_ + s];   // a_self_rev
  const int t_of_t = targets[b * E_ + t];   // a_adjc_rev
  const size_t nb = (size_t)b * NN_;
  const float* r0 = AS + (nb + t)      * H_;
  const float* r1 = AA + (nb + s)      * H_;
  const float* r2 = AS + (nb + t_of_s) * H_;
  const float* r3 = AS + (nb + t_of_t) * H_;
#pragma unroll
  for (int h = 0; h < H_; ++h) {
    float x = r0[h] + r1[h] + r2[h] + r3[h];
    v[h] = x > 0.f ? x : 0.2f * x;          // leaky_relu(0.2)
  }
  return t;
}

// ---------------------------------------------------------------------------
// Phase A: first segment-max (batches merged: seg = targets.reshape(-1)).
// ---------------------------------------------------------------------------
__global__ void k_segmax1(const float* __restrict__ AS, const float* __restrict__ AA,
                          const int* __restrict__ targets, const int* __restrict__ sources,
                          float* __restrict__ m) {
  int i = blockIdx.x * blockDim.x + threadIdx.x;
  if (i >= B_ * E_) return;
  float v[H_];
  const int t = gather_coef(AS, AA, targets, sources, i, v);
  float* mrow = m + (size_t)t * H_;
#pragma unroll
  for (int h = 0; h < H_; ++h) atomicMax(mrow + h, v[h]);
}

// ---------------------------------------------------------------------------
// Phase B: c = exp(v - m[t]); second segment-max into m2 (no coef store).
// ---------------------------------------------------------------------------
__global__ void k_segmax2(const float* __restrict__ AS, const float* __restrict__ AA,
                          const int* __restrict__ targets, const int* __restrict__ sources,
                          const float* __restrict__ m, float* __restrict__ m2) {
  int i = blockIdx.x * blockDim.x + threadIdx.x;
  if (i >= B_ * E_) return;
  float v[H_];
  const int t = gather_coef(AS, AA, targets, sources, i, v);
  const float* mrow = m + (size_t)t * H_;
  float* m2row = m2 + (size_t)t * H_;
#pragma unroll
  for (int h = 0; h < H_; ++h) atomicMax(m2row + h, expf(v[h] - mrow[h]));
}

// ---------------------------------------------------------------------------
// Threefry2x32 (Random123, 20 rounds) matching jax.random.key(1234):
// key = (0, 1234). Counter split per JAX threefry_random_bits.
// ---------------------------------------------------------------------------
__device__ __forceinline__ unsigned tf_rotl(unsigned x, int n) {
  return (x << n) | (x >> (32 - n));
}
__device__ __forceinline__ unsigned threefry2x32(unsigned c0, unsigned c1, int which) {
  const unsigned k0 = 0u, k1 = 1234u;
  const unsigned k2 = 0x1BD11BDAu ^ k0 ^ k1;
  unsigned x0 = c0 + k0, x1 = c1 + k1;
#define TF_G(a, b, cc, d)                                   \
  x0 += x1; x1 = tf_rotl(x1, a); x1 ^= x0;                  \
  x0 += x1; x1 = tf_rotl(x1, b); x1 ^= x0;                  \
  x0 += x1; x1 = tf_rotl(x1, cc); x1 ^= x0;                 \
  x0 += x1; x1 = tf_rotl(x1, d); x1 ^= x0;
  TF_G(13, 15, 26, 6)  x0 += k1; x1 += k2 + 1u;
  TF_G(17, 29, 16, 24) x0 += k2; x1 += k0 + 2u;
  TF_G(13, 15, 26, 6)  x0 += k0; x1 += k1 + 3u;
  TF_G(17, 29, 16, 24) x0 += k1; x1 += k2 + 4u;
  TF_G(13, 15, 26, 6)  x0 += k2; x1 += k0 + 5u;
#undef TF_G
  return which ? x1 : x0;
}

// ---------------------------------------------------------------------------
// Phase C: recompute v and c, out = dropout( c / (m2[t] + 1e-9) ), p=0.5.
// JAX uniform: f = bitcast((bits >> 9) | 0x3F800000) - 1; keep = f < 0.5.
// Single 51.2 MB HBM write; everything else is L2-resident.
// ---------------------------------------------------------------------------
__global__ void k_final(const float* __restrict__ AS, const float* __restrict__ AA,
                        const int* __restrict__ targets, const int* __restrict__ sources,
                        const float* __restrict__ m, const float* __restrict__ m2,
                        float* __restrict__ out) {
  int i = blockIdx.x * blockDim.x + threadIdx.x;
  if (i >= B_ * E_) return;
  float v[H_];
  const int t = gather_coef(AS, AA, targets, sources, i, v);
  const float* mrow  = m  + (size_t)t * H_;
  const float* m2row = m2 + (size_t)t * H_;
  float* orow = out + (size_t)i * H_;
  const unsigned n2 = (unsigned)(B_) * E_ * H_ / 2u;
#pragma unroll
  for (int h = 0; h < H_; ++h) {
    const unsigned idx = (unsigned)i * H_ + h;
    const int which = idx >= n2;
    const unsigned j = which ? idx - n2 : idx;
    const unsigned bits = threefry2x32(j, j + n2, which);
    const float u = __uint_as_float((bits >> 9) | 0x3F800000u) - 1.0f;
    const float c = expf(v[h] - mrow[h]);
    const float r = c / (m2row[h] + 1e-9f);
    orow[h] = (u < 0.5f) ? r * 2.0f : 0.0f;
  }
}

// ---------------------------------------------------------------------------
// d_in order: X, N, targets, sources, degree, attn_kernel_self, attn_kernel_adjc
// (N and degree are unused by the reference). Workspace layout (floats):
// AS[B*NN*H] | AA[B*NN*H] | m[NN*H] | m2[NN*H]   (9.6 MB total)
// ---------------------------------------------------------------------------
extern "C" void kernel_launch(void* const* d_in, const int* in_sizes, int n_in,
                              void* d_out, int out_size, void* d_ws, size_t ws_size,
                              hipStream_t stream) {
  const float* X       = (const float*)d_in[0];
  const int*   targets = (const int*)d_in[2];
  const int*   sources = (const int*)d_in[3];
  const float* ks      = (const float*)d_in[5];
  const float* ka      = (const float*)d_in[6];

  float* ws = (float*)d_ws;
  float* AS = ws;
  float* AA = AS + (size_t)B_ * NN_ * H_;
  float* m  = AA + (size_t)B_ * NN_ * H_;
  float* m2 = m  + (size_t)NN_ * H_;
  float* out = (float*)d_out;

  {
    const int n = 2 * NN_ * H_;                 // m and m2 are contiguous
    k_init<<<(n + 255) / 256, 256, 0, stream>>>(m, n);
  }
  {
    const int numTiles = B_ * NN_ / 2;          // 50000 tiles of 16 rows
    k_attn_scores<<<1024, 256, 0, stream>>>(X, ks, ka, AS, AA, numTiles);
  }
  const int nbe = B_ * E_;
  k_segmax1<<<(nbe + 255) / 256, 256, 0, stream>>>(AS, AA, targets, sources, m);
  k_segmax2<<<(nbe + 255) / 256, 256, 0, stream>>>(AS, AA, targets, sources, m, m2);
  k_final  <<<(nbe + 255) / 256, 256, 0, stream>>>(AS, AA, targets, sources, m, m2, out);
}